// GINAttention_52956946760187
// MI455X (gfx1250) — compile-verified
//
#include <hip/hip_runtime.h>
#include <math.h>

typedef __attribute__((ext_vector_type(2))) float v2f;
typedef __attribute__((ext_vector_type(8))) float v8f;

#define HID 128
#define BN_EPS 1e-5f

// ---------------------------------------------------------------------------
// Generic helpers
// ---------------------------------------------------------------------------
__global__ __launch_bounds__(256) void k_copy(const float* __restrict__ src,
                                              float* __restrict__ dst, int n) {
  int i = blockIdx.x * 256 + threadIdx.x;
  if (i < n) dst[i] = src[i];
}

__global__ __launch_bounds__(256) void k_zero(float* __restrict__ dst, int n) {
  int i = blockIdx.x * 256 + threadIdx.x;
  if (i < n) dst[i] = 0.0f;
}

// One-time 128x128 weight transpose: Wt[n][k] = W[k][n]
__global__ __launch_bounds__(256) void k_transpose128(const float* __restrict__ W,
                                                      float* __restrict__ Wt) {
  int i = blockIdx.x * 256 + threadIdx.x;
  if (i >= HID * HID) return;
  int n = i >> 7, k = i & (HID - 1);
  Wt[n * HID + k] = W[k * HID + n];
}

// ---------------------------------------------------------------------------
// Edge feature scatter-add:  dst[col[e]] += src[row[e]]  (128 feats per edge)
// One wave (32 lanes) per edge; each lane moves 4 contiguous floats.
// ---------------------------------------------------------------------------
__global__ __launch_bounds__(256) void k_scatter_add_feat(
    const float* __restrict__ src, const int* __restrict__ row,
    const int* __restrict__ col, float* __restrict__ dst, int E) {
  int wave = (blockIdx.x * 256 + threadIdx.x) >> 5;
  int lane = threadIdx.x & 31;
  if (wave >= E) return;
  int r = row[wave];
  int c = col[wave];
  const float4 v = *(const float4*)(src + (size_t)r * HID + lane * 4);
  float* d = dst + (size_t)c * HID + lane * 4;
  atomicAdd(d + 0, v.x);
  atomicAdd(d + 1, v.y);
  atomicAdd(d + 2, v.z);
  atomicAdd(d + 3, v.w);
}

// ---------------------------------------------------------------------------
// WMMA fp32 GEMM: C[M,128] = relu?( bn_relu?(A)[M,128] @ Wt^T + bias )
//  * Wt is the PRE-TRANSPOSED weight: Wt[n][k] = W[k][n]  -> b64 B-frag loads
//  * Block = 256 threads = 8 waves; block owns 32 rows (two 16-row subtiles),
//    wave w owns output cols 16w..16w+15. Each B-fragment feeds 2 WMMAs.
//  * Optional fused input transform: a = relu(a*in_scale[c] + in_shift[c])
//    (folded BatchNorm affine), applied while staging the A tile in LDS.
//  * Per wave: 64 x v_wmma_f32_16x16x4_f32 over two accumulator chains.
// ---------------------------------------------------------------------------
__global__ __launch_bounds__(256) void k_gemm128_wmma(
    const float* __restrict__ A, const float* __restrict__ Wt,
    const float* __restrict__ bias, float* __restrict__ C, int M,
    const float* __restrict__ in_scale, const float* __restrict__ in_shift,
    int relu_out) {
  __shared__ float Atile[32 * HID];

  const int m0 = blockIdx.x * 32;
  // Stage 32x128 A tile: 1024 float4 by 256 threads; zero-pad rows >= M.
  for (int i = threadIdx.x; i < 1024; i += 256) {
    int r = i >> 5;             // tile row (32 float4 per row)
    int c4 = (i & 31) * 4;      // starting column of this float4
    float4 v = make_float4(0.f, 0.f, 0.f, 0.f);
    if (m0 + r < M) {
      v = *(const float4*)(A + (size_t)(m0 + r) * HID + c4);
      if (in_scale) {  // fused BN + ReLU on the GEMM input
        v.x = fmaxf(v.x * in_scale[c4 + 0] + in_shift[c4 + 0], 0.f);
        v.y = fmaxf(v.y * in_scale[c4 + 1] + in_shift[c4 + 1], 0.f);
        v.z = fmaxf(v.z * in_scale[c4 + 2] + in_shift[c4 + 2], 0.f);
        v.w = fmaxf(v.w * in_scale[c4 + 3] + in_shift[c4 + 3], 0.f);
      }
    }
    *(float4*)(Atile + r * HID + c4) = v;
  }
  __syncthreads();

  const int wave = threadIdx.x >> 5;
  const int lane = threadIdx.x & 31;
  const int half = lane >> 4;  // 0: K pair {0,1}, 1: K pair {2,3}
  const int l15  = lane & 15;
  const int n0   = wave * 16;
  const float* WtRow = Wt + (size_t)(n0 + l15) * HID;  // lane's B column

  v8f acc0 = {0.f, 0.f, 0.f, 0.f, 0.f, 0.f, 0.f, 0.f};
  v8f acc1 = {0.f, 0.f, 0.f, 0.f, 0.f, 0.f, 0.f, 0.f};
#pragma unroll
  for (int k = 0; k < HID; k += 4) {
    // A frags: lane (m=l15) holds A[m][k+2h], A[m][k+2h+1]  (two subtiles)
    v2f a0, a1;
    a0.x = Atile[l15 * HID + k + half * 2 + 0];
    a0.y = Atile[l15 * HID + k + half * 2 + 1];
    a1.x = Atile[(16 + l15) * HID + k + half * 2 + 0];
    a1.y = Atile[(16 + l15) * HID + k + half * 2 + 1];
    // B frag: lane (n=l15) holds W[k+2h][n], W[k+2h+1][n] == contiguous in Wt
    const v2f b = *(const v2f*)(WtRow + k + half * 2);
    acc0 = __builtin_amdgcn_wmma_f32_16x16x4_f32(false, a0, false, b,
                                                 (short)0, acc0, false, false);
    acc1 = __builtin_amdgcn_wmma_f32_16x16x4_f32(false, a1, false, b,
                                                 (short)0, acc1, false, false);
  }

  const float bv = bias[n0 + l15];
#pragma unroll
  for (int r = 0; r < 8; ++r) {
    // C VGPR r: row = r + 8*half (within subtile), col = l15
    int row0 = m0 + half * 8 + r;
    float v0 = acc0[r] + bv;
    if (relu_out) v0 = fmaxf(v0, 0.0f);
    if (row0 < M) C[(size_t)row0 * HID + n0 + l15] = v0;
    int row1 = row0 + 16;
    float v1 = acc1[r] + bv;
    if (relu_out) v1 = fmaxf(v1, 0.0f);
    if (row1 < M) C[(size_t)row1 * HID + n0 + l15] = v1;
  }
}

// ---------------------------------------------------------------------------
// BatchNorm batch statistics folded to affine form:
//   scale[c] = g[c] * rsqrt(var + eps),  shift[c] = be[c] - mean * scale[c]
// One block per column (128 blocks), 256 threads stride the M rows.
// ---------------------------------------------------------------------------
__global__ __launch_bounds__(256) void k_bn_stats_affine(
    const float* __restrict__ Z, int M, const float* __restrict__ g,
    const float* __restrict__ be, float* __restrict__ scale,
    float* __restrict__ shift) {
  __shared__ float ssum[256];
  __shared__ float ssq[256];
  const int c = blockIdx.x;
  float s = 0.f, q = 0.f;
  for (int i = threadIdx.x; i < M; i += 256) {
    float x = Z[(size_t)i * HID + c];
    s += x;
    q += x * x;
  }
  ssum[threadIdx.x] = s;
  ssq[threadIdx.x] = q;
  __syncthreads();
  for (int off = 128; off > 0; off >>= 1) {
    if (threadIdx.x < off) {
      ssum[threadIdx.x] += ssum[threadIdx.x + off];
      ssq[threadIdx.x] += ssq[threadIdx.x + off];
    }
    __syncthreads();
  }
  if (threadIdx.x == 0) {
    float m = ssum[0] / (float)M;
    float v = ssq[0] / (float)M - m * m;
    float sc = g[c] * rsqrtf(v + BN_EPS);
    scale[c] = sc;
    shift[c] = be[c] - m * sc;
  }
}

// ---------------------------------------------------------------------------
// Attention pool, exploiting gather==scatter index:
//   s[i] = sigmoid(H[i]·aw + ab);  t[i] = sum_{e: col_e==i} s[row_e];
//   out[i] = s[i] * t[i] * H[i]
// ---------------------------------------------------------------------------
__global__ __launch_bounds__(256) void k_attn_score(
    const float* __restrict__ H, const float* __restrict__ aw,
    const float* __restrict__ ab, float* __restrict__ s, int N) {
  int wave = (blockIdx.x * 256 + threadIdx.x) >> 5;
  int lane = threadIdx.x & 31;
  if (wave >= N) return;
  const float4 h = *(const float4*)(H + (size_t)wave * HID + lane * 4);
  const float4 w = *(const float4*)(aw + lane * 4);
  float d = h.x * w.x + h.y * w.y + h.z * w.z + h.w * w.w;
#pragma unroll
  for (int off = 16; off > 0; off >>= 1) d += __shfl_xor(d, off, 32);
  if (lane == 0) s[wave] = 1.0f / (1.0f + expf(-(d + ab[0])));
}

__global__ __launch_bounds__(256) void k_attn_accum(
    const float* __restrict__ s, const int* __restrict__ row,
    const int* __restrict__ col, float* __restrict__ t, int E) {
  int e = blockIdx.x * 256 + threadIdx.x;
  if (e >= E) return;
  atomicAdd(&t[col[e]], s[row[e]]);
}

__global__ __launch_bounds__(256) void k_attn_apply(
    const float* __restrict__ H, const float* __restrict__ s,
    const float* __restrict__ t, float* __restrict__ out, int n) {
  int i = blockIdx.x * 256 + threadIdx.x;
  if (i >= n) return;
  int node = i >> 7;
  out[i] = H[i] * s[node] * t[node];
}

// hg[batch[i]] += H[i]  (per (node,feature) thread)
__global__ __launch_bounds__(256) void k_graph_pool(
    const float* __restrict__ H, const int* __restrict__ batch,
    float* __restrict__ hg, int n) {
  int i = blockIdx.x * 256 + threadIdx.x;
  if (i >= n) return;
  int node = i >> 7;
  int f = i & (HID - 1);
  atomicAdd(&hg[(size_t)batch[node] * HID + f], H[i]);
}

// Small head GEMM with fused BN+ReLU input: out[G,10] = relu(bn(A)) @ W + b
__global__ __launch_bounds__(256) void k_head_gemm(
    const float* __restrict__ A, const float* __restrict__ sc,
    const float* __restrict__ sh, const float* __restrict__ W,
    const float* __restrict__ b, float* __restrict__ out, int G) {
  int i = blockIdx.x * 256 + threadIdx.x;
  if (i >= G * 10) return;
  int g = i / 10, o = i % 10;
  float acc = b[o];
  for (int k = 0; k < HID; ++k) {
    float a = fmaxf(A[(size_t)g * HID + k] * sc[k] + sh[k], 0.0f);
    acc += a * W[k * 10 + o];
  }
  out[i] = acc;
}

__global__ __launch_bounds__(64) void k_log_softmax10(
    const float* __restrict__ in, float* __restrict__ out, int G) {
  int g = blockIdx.x * 64 + threadIdx.x;
  if (g >= G) return;
  float m = in[g * 10];
  for (int o = 1; o < 10; ++o) m = fmaxf(m, in[g * 10 + o]);
  float s = 0.f;
  for (int o = 0; o < 10; ++o) s += expf(in[g * 10 + o] - m);
  float ls = logf(s);
  for (int o = 0; o < 10; ++o) out[g * 10 + o] = in[g * 10 + o] - m - ls;
}

// ---------------------------------------------------------------------------
// Orchestration
// ---------------------------------------------------------------------------
extern "C" void kernel_launch(void* const* d_in, const int* in_sizes, int n_in,
                              void* d_out, int out_size, void* d_ws, size_t ws_size,
                              hipStream_t stream) {
  const float* x      = (const float*)d_in[0];
  const int*   edge   = (const int*)d_in[1];
  const int*   batch  = (const int*)d_in[2];
  const float* c1_w1 = (const float*)d_in[3],  *c1_b1 = (const float*)d_in[4];
  const float* c1_g  = (const float*)d_in[5],  *c1_be = (const float*)d_in[6];
  const float* c1_w2 = (const float*)d_in[7],  *c1_b2 = (const float*)d_in[8];
  const float* a1_w  = (const float*)d_in[9],  *a1_b  = (const float*)d_in[10];
  const float* c2_w1 = (const float*)d_in[11], *c2_b1 = (const float*)d_in[12];
  const float* c2_g  = (const float*)d_in[13], *c2_be = (const float*)d_in[14];
  const float* c2_w2 = (const float*)d_in[15], *c2_b2 = (const float*)d_in[16];
  const float* a2_w  = (const float*)d_in[17], *a2_b  = (const float*)d_in[18];
  const float* l_w1  = (const float*)d_in[19], *l_b1  = (const float*)d_in[20];
  const float* l_g   = (const float*)d_in[21], *l_be  = (const float*)d_in[22];
  const float* l_w2  = (const float*)d_in[23], *l_b2  = (const float*)d_in[24];

  const int N = in_sizes[0] / HID;      // 10000
  const int E = in_sizes[1] / 2;        // 640000
  const int G = out_size / 10;          // 64
  const int* row = edge;
  const int* col = edge + E;

  // Workspace carve-up (floats)
  float* B0  = (float*)d_ws;                 // [N,128]
  float* B1  = B0 + (size_t)N * HID;         // [N,128]
  float* B2  = B1 + (size_t)N * HID;         // [N,128]
  float* S   = B2 + (size_t)N * HID;         // [N]
  float* T   = S + N;                        // [N]
  float* SC  = T + N;                        // [128] folded BN scale
  float* SH  = SC + HID;                     // [128] folded BN shift
  float* HG  = SH + HID;                     // [G,128]
  float* Z2  = HG + (size_t)G * HID;         // [G,128]
  float* O10 = Z2 + (size_t)G * HID;         // [G,10]
  float* WT1 = O10 + (size_t)G * 10;         // 5 x [128,128] transposed weights
  float* WT2 = WT1 + HID * HID;
  float* WT3 = WT2 + HID * HID;
  float* WT4 = WT3 + HID * HID;
  float* WT5 = WT4 + HID * HID;

  const int nf = N * HID;
  dim3 blk(256);
  const int g_nf    = (nf + 255) / 256;
  const int g_edge8 = (E + 7) / 8;           // wave per edge
  const int g_edge  = (E + 255) / 256;
  const int g_rows8 = (N + 7) / 8;           // wave per node
  const int g_gemmN = (N + 31) / 32;         // 32-row GEMM blocks
  const int g_gemmG = (G + 31) / 32;
  const int g_tr    = (HID * HID + 255) / 256;

  // ---- one-time weight transposes (Wt[n][k] = W[k][n]) ----
  k_transpose128<<<g_tr, blk, 0, stream>>>(c1_w1, WT1);
  k_transpose128<<<g_tr, blk, 0, stream>>>(c1_w2, WT2);
  k_transpose128<<<g_tr, blk, 0, stream>>>(c2_w1, WT3);
  k_transpose128<<<g_tr, blk, 0, stream>>>(c2_w2, WT4);
  k_transpose128<<<g_tr, blk, 0, stream>>>(l_w1,  WT5);

  // ---- GIN conv 1: B0 = x + segsum(x[row]->col); MLP ----
  k_copy<<<g_nf, blk, 0, stream>>>(x, B0, nf);
  k_scatter_add_feat<<<g_edge8, blk, 0, stream>>>(x, row, col, B0, E);
  k_gemm128_wmma<<<g_gemmN, blk, 0, stream>>>(B0, WT1, c1_b1, B1, N,
                                              nullptr, nullptr, 0);        // z1
  k_bn_stats_affine<<<HID, blk, 0, stream>>>(B1, N, c1_g, c1_be, SC, SH);
  k_gemm128_wmma<<<g_gemmN, blk, 0, stream>>>(B1, WT2, c1_b2, B0, N,
                                              SC, SH, 1);                  // h1

  // ---- attn pool 1 ----
  k_attn_score<<<g_rows8, blk, 0, stream>>>(B0, a1_w, a1_b, S, N);
  k_zero<<<(N + 255) / 256, blk, 0, stream>>>(T, N);
  k_attn_accum<<<g_edge, blk, 0, stream>>>(S, row, col, T, E);
  k_attn_apply<<<g_nf, blk, 0, stream>>>(B0, S, T, B1, nf);                // h

  // ---- GIN conv 2 ----
  k_copy<<<g_nf, blk, 0, stream>>>(B1, B2, nf);
  k_scatter_add_feat<<<g_edge8, blk, 0, stream>>>(B1, row, col, B2, E);
  k_gemm128_wmma<<<g_gemmN, blk, 0, stream>>>(B2, WT3, c2_b1, B0, N,
                                              nullptr, nullptr, 0);        // z2
  k_bn_stats_affine<<<HID, blk, 0, stream>>>(B0, N, c2_g, c2_be, SC, SH);
  k_gemm128_wmma<<<g_gemmN, blk, 0, stream>>>(B0, WT4, c2_b2, B2, N,
                                              SC, SH, 1);                  // h2

  // ---- attn pool 2 ----
  k_attn_score<<<g_rows8, blk, 0, stream>>>(B2, a2_w, a2_b, S, N);
  k_zero<<<(N + 255) / 256, blk, 0, stream>>>(T, N);
  k_attn_accum<<<g_edge, blk, 0, stream>>>(S, row, col, T, E);
  k_attn_apply<<<g_nf, blk, 0, stream>>>(B2, S, T, B1, nf);                // h

  // ---- graph pooling + head MLP ----
  k_zero<<<(G * HID + 255) / 256, blk, 0, stream>>>(HG, G * HID);
  k_graph_pool<<<g_nf, blk, 0, stream>>>(B1, batch, HG, nf);
  k_gemm128_wmma<<<g_gemmG, blk, 0, stream>>>(HG, WT5, l_b1, Z2, G,
                                              nullptr, nullptr, 0);
  k_bn_stats_affine<<<HID, blk, 0, stream>>>(Z2, G, l_g, l_be, SC, SH);
  k_head_gemm<<<(G * 10 + 255) / 256, blk, 0, stream>>>(Z2, SC, SH, l_w2,
                                                        l_b2, O10, G);
  k_log_softmax10<<<(G + 63) / 64, dim3(64), 0, stream>>>(O10, (float*)d_out, G);
}